// LongPoly_88630945120296
// MI455X (gfx1250) — compile-verified
//
#include <hip/hip_runtime.h>
#include <hip/hip_bf16.h>

// ---------------------------------------------------------------------------
// ChebNet-style graph conv for MI455X (gfx1250, wave32).
// Memory-bound: 5x sparse gather/scatter (~2 GB touched, mostly L2-resident
// since each 51.2MB operand fits in the 192MB L2). Epilogue RMSNorm uses
// V_WMMA_F32_16X16X4_F32 (A = squared features, B = ones) to reduce 16 node
// rows per wave.
// ---------------------------------------------------------------------------

#define NN   100000
#define EE   400000
#define HH   128
#define NH   (NN * HH)         // 12,800,000 floats
#define NH4  (NH / 4)          // float4 chunks
#define KP1  6                 // K+1 coefficients per group
#define RMS_EPS 1.1920929e-07f

typedef __attribute__((ext_vector_type(2))) float v2f;
typedef __attribute__((ext_vector_type(8))) float v8f;

__device__ __forceinline__ float silu_f(float v) {
  return v / (1.0f + __expf(-v));
}

// ---------------------------------------------------------------------------
// acc = c0[g] * x           (one float4 per thread, g = channel group)
// ---------------------------------------------------------------------------
__global__ __launch_bounds__(256) void cheb_init_kernel(
    const float* __restrict__ x, const float* __restrict__ coeffs,
    float* __restrict__ acc) {
  int idx = blockIdx.x * blockDim.x + threadIdx.x;   // float4 index
  if (idx >= NH4) return;
  int g = (idx & 31) >> 3;                           // (col4*4)/32
  float c0 = coeffs[g * KP1 + 0];
  float4 v = ((const float4*)x)[idx];
  float4 o = make_float4(v.x * c0, v.y * c0, v.z * c0, v.w * c0);
  ((float4*)acc)[idx] = o;
}

// ---------------------------------------------------------------------------
// acc += c_k[g] * T
// ---------------------------------------------------------------------------
__global__ __launch_bounds__(256) void cheb_acc_kernel(
    const float* __restrict__ T, const float* __restrict__ coeffs, int k,
    float* __restrict__ acc) {
  int idx = blockIdx.x * blockDim.x + threadIdx.x;
  if (idx >= NH4) return;
  int g = (idx & 31) >> 3;
  float ck = coeffs[g * KP1 + k];
  float4 t = ((const float4*)T)[idx];
  float4 a = ((float4*)acc)[idx];
  a.x += ck * t.x; a.y += ck * t.y; a.z += ck * t.z; a.w += ck * t.w;
  ((float4*)acc)[idx] = a;
}

// ---------------------------------------------------------------------------
// Tc = 2*Tc - Tb;  acc += c_k[g] * Tc        (Chebyshev recurrence + combine)
// ---------------------------------------------------------------------------
__global__ __launch_bounds__(256) void cheb_step_kernel(
    float* __restrict__ Tc, const float* __restrict__ Tb,
    const float* __restrict__ coeffs, int k, float* __restrict__ acc) {
  int idx = blockIdx.x * blockDim.x + threadIdx.x;
  if (idx >= NH4) return;
  int g = (idx & 31) >> 3;
  float ck = coeffs[g * KP1 + k];
  float4 tc = ((const float4*)Tc)[idx];
  float4 tb = ((const float4*)Tb)[idx];
  float4 t  = make_float4(2.0f * tc.x - tb.x, 2.0f * tc.y - tb.y,
                          2.0f * tc.z - tb.z, 2.0f * tc.w - tb.w);
  ((float4*)Tc)[idx] = t;
  float4 a = ((float4*)acc)[idx];
  a.x += ck * t.x; a.y += ck * t.y; a.z += ck * t.z; a.w += ck * t.w;
  ((float4*)acc)[idx] = a;
}

// ---------------------------------------------------------------------------
// Sparse propagate: out[dst] += h[src] * ew.  One wave32 per edge:
// 32 lanes x float4 = 512B row; 4 hardware f32 atomic adds per lane (L2 RMW).
// ---------------------------------------------------------------------------
__global__ __launch_bounds__(256) void prop_kernel(
    const float* __restrict__ h, const int* __restrict__ ei,
    const float* __restrict__ ew, float* __restrict__ out) {
  int wid  = (blockIdx.x * blockDim.x + threadIdx.x) >> 5;  // edge id
  if (wid >= EE) return;
  int lane = threadIdx.x & 31;
  int s = ei[wid];        // edge_index[0][e]
  int d = ei[EE + wid];   // edge_index[1][e]
  float w = ew[wid];
  float4 v = ((const float4*)(h + (size_t)s * HH))[lane];
  float* o = out + (size_t)d * HH + lane * 4;
  unsafeAtomicAdd(o + 0, v.x * w);
  unsafeAtomicAdd(o + 1, v.y * w);
  unsafeAtomicAdd(o + 2, v.z * w);
  unsafeAtomicAdd(o + 3, v.w * w);
}

// ---------------------------------------------------------------------------
// Epilogue: y = acc*scale[g]+bias[g]; rms over H via WMMA f32 16x16x4
// (A = y^2 tile, B = ones -> D[m][*] = row sum); out = silu(y*rms*w[h]).
// 128 threads = 4 waves, each wave owns 16 node rows.
// LDS rows padded to 132 floats (132 % 64 == 4) so the strided A-matrix
// column reads (lane = M) hit 16 distinct banks.
// ---------------------------------------------------------------------------
#define ROWPAD 132
__global__ __launch_bounds__(128) void epilogue_kernel(
    const float* __restrict__ acc, const float* __restrict__ gscale,
    const float* __restrict__ gbias, const float* __restrict__ rmsw,
    float* __restrict__ out) {
  __shared__ float ys[4][16 * ROWPAD];
  __shared__ float rsums[4][16];
  const int w    = threadIdx.x >> 5;
  const int lane = threadIdx.x & 31;
  const int rowbase = blockIdx.x * 64 + w * 16;

  // ---- phase 1: stage y = acc*scale+bias into LDS (zeros past N) ----
  const int g = lane >> 3;               // chunk col = lane*4 -> group
  const float sc = gscale[g];
  const float bi = gbias[g];
  #pragma unroll 4
  for (int i = 0; i < 16; ++i) {
    int row = rowbase + i;
    float4 y = make_float4(0.f, 0.f, 0.f, 0.f);
    if (row < NN) {
      float4 a = ((const float4*)(acc + (size_t)row * HH))[lane];
      y = make_float4(a.x * sc + bi, a.y * sc + bi,
                      a.z * sc + bi, a.w * sc + bi);
    }
    float* p = &ys[w][i * ROWPAD + lane * 4];
    p[0] = y.x; p[1] = y.y; p[2] = y.z; p[3] = y.w;
  }
  __syncthreads();

  // ---- phase 2: row sum-of-squares via V_WMMA_F32_16X16X4_F32 ----
  // A layout (16x4 f32): lane L -> M = L&15; VGPR0/1 = K = k0 + 2*(L>=16) + {0,1}
  v8f c = {};
  v2f b; b.x = 1.0f; b.y = 1.0f;                 // B = ones (layout-invariant)
  const int m = lane & 15;
  const int koff = (lane >> 4) << 1;             // 0 or 2
  #pragma unroll
  for (int k0 = 0; k0 < HH; k0 += 4) {
    float2 av = *(const float2*)&ys[w][m * ROWPAD + k0 + koff];
    v2f a; a.x = av.x * av.x; a.y = av.y * av.y;
    c = __builtin_amdgcn_wmma_f32_16x16x4_f32(
        /*neg_a=*/false, a, /*neg_b=*/false, b,
        /*c_mod=*/(short)0, c, /*reuse_a=*/false, /*reuse_b=*/false);
  }
  // D layout: VGPR j holds row j (lanes 0-15) / row j+8 (lanes 16-31).
  if ((lane & 15) == 0) {
    const int rbase = (lane >> 4) << 3;          // 0 or 8
    #pragma unroll
    for (int j = 0; j < 8; ++j) rsums[w][rbase + j] = c[j];
  }
  __syncthreads();

  // ---- phase 3: rms + weight + silu, write out ----
  const float4 rw = ((const float4*)rmsw)[lane];
  #pragma unroll 4
  for (int i = 0; i < 16; ++i) {
    int row = rowbase + i;
    if (row >= NN) continue;
    float r = rsqrtf(rsums[w][i] * (1.0f / (float)HH) + RMS_EPS);
    const float* p = &ys[w][i * ROWPAD + lane * 4];
    float4 o;
    o.x = silu_f(p[0] * r * rw.x);
    o.y = silu_f(p[1] * r * rw.y);
    o.z = silu_f(p[2] * r * rw.z);
    o.w = silu_f(p[3] * r * rw.w);
    ((float4*)(out + (size_t)row * HH))[lane] = o;
  }
}

// ---------------------------------------------------------------------------
extern "C" void kernel_launch(void* const* d_in, const int* in_sizes, int n_in,
                              void* d_out, int out_size, void* d_ws, size_t ws_size,
                              hipStream_t stream) {
  const float* x      = (const float*)d_in[0];   // [N,H]
  const float* ew     = (const float*)d_in[1];   // [E]
  const float* cheb   = (const float*)d_in[2];   // [4,6]
  const float* gscale = (const float*)d_in[3];   // [4]
  const float* gbias  = (const float*)d_in[4];   // [4]
  const float* rmsw   = (const float*)d_in[5];   // [128]
  const int*   ei     = (const int*)d_in[6];     // [2,E]
  float* out = (float*)d_out;

  float* ws  = (float*)d_ws;
  float* acc = ws;                         // result accumulator
  float* B0  = ws + (size_t)NH;            // rotating T buffers
  float* B1  = ws + 2 * (size_t)NH;
  float* B2  = ws + 3 * (size_t)NH;
  const size_t tbytes = (size_t)NH * sizeof(float);

  const int threads = 256;
  const int gridNH  = (NH4 + threads - 1) / threads;
  const int gridE   = (int)(((size_t)EE * 32 + threads - 1) / threads);
  const int gridEp  = (NN + 63) / 64;

  // acc = c0 * x
  cheb_init_kernel<<<gridNH, threads, 0, stream>>>(x, cheb, acc);

  // T1 = A x ; acc += c1 * T1
  hipMemsetAsync(B0, 0, tbytes, stream);
  prop_kernel<<<gridE, threads, 0, stream>>>(x, ei, ew, B0);
  cheb_acc_kernel<<<gridNH, threads, 0, stream>>>(B0, cheb, 1, acc);

  // k = 2: Tn=B1 = 2*A*B0 - x
  hipMemsetAsync(B1, 0, tbytes, stream);
  prop_kernel<<<gridE, threads, 0, stream>>>(B0, ei, ew, B1);
  cheb_step_kernel<<<gridNH, threads, 0, stream>>>(B1, x, cheb, 2, acc);

  // k = 3: Tn=B2 = 2*A*B1 - B0
  hipMemsetAsync(B2, 0, tbytes, stream);
  prop_kernel<<<gridE, threads, 0, stream>>>(B1, ei, ew, B2);
  cheb_step_kernel<<<gridNH, threads, 0, stream>>>(B2, B0, cheb, 3, acc);

  // k = 4: Tn=B0 = 2*A*B2 - B1
  hipMemsetAsync(B0, 0, tbytes, stream);
  prop_kernel<<<gridE, threads, 0, stream>>>(B2, ei, ew, B0);
  cheb_step_kernel<<<gridNH, threads, 0, stream>>>(B0, B1, cheb, 4, acc);

  // k = 5: Tn=B1 = 2*A*B0 - B2
  hipMemsetAsync(B1, 0, tbytes, stream);
  prop_kernel<<<gridE, threads, 0, stream>>>(B0, ei, ew, B1);
  cheb_step_kernel<<<gridNH, threads, 0, stream>>>(B1, B2, cheb, 5, acc);

  // scale/bias + RMSNorm(WMMA) + SiLU
  epilogue_kernel<<<gridEp, 128, 0, stream>>>(acc, gscale, gbias, rmsw, out);
}